// TripletLoss_88089779241009
// MI455X (gfx1250) — compile-verified
//
#include <hip/hip_runtime.h>

typedef float v2f __attribute__((ext_vector_type(2)));
typedef float v8f __attribute__((ext_vector_type(8)));

#define NROW 4096
#define NDIM 2048
#define MARGIN 0.3f
#define POS_INF_BITS 0x7F800000

// ---------------------------------------------------------------------------
// Kernel 1: per-row squared norms + init of per-row max/min accumulators.
// ---------------------------------------------------------------------------
__global__ void prep_kernel(const float* __restrict__ X,
                            float* __restrict__ sq,
                            float* __restrict__ d_ap,
                            float* __restrict__ d_an) {
    const int row = blockIdx.x;
    const float* xr = X + (size_t)row * NDIM;
    float s = 0.0f;
    for (int k = threadIdx.x; k < NDIM; k += 256) {
        float v = xr[k];
        s += v * v;
    }
    __shared__ float red[256];
    red[threadIdx.x] = s;
    __syncthreads();
    for (int off = 128; off > 0; off >>= 1) {
        if (threadIdx.x < off) red[threadIdx.x] += red[threadIdx.x + off];
        __syncthreads();
    }
    if (threadIdx.x == 0) {
        sq[row]   = red[0];
        d_ap[row] = 0.0f;                          // diag positive guarantees >= 0
        d_an[row] = __int_as_float(POS_INF_BITS);  // +inf
    }
}

// ---------------------------------------------------------------------------
// Kernel 2: fused Gram-tile (fp32 WMMA 16x16x4) + hard-mining epilogue.
// One wave per block computes a 64x64 tile of G = X X^T, upper triangle only
// (bx >= by); symmetry lets each tile update both its rows and its columns.
// A-tile lane layout == B-tile lane layout == rows of X:
//   lane (sub,half): element (idx=sub, k = 2*half + vgpr)
// ---------------------------------------------------------------------------
__global__ __launch_bounds__(32) void gram_mine_kernel(
        const float* __restrict__ X,
        const float* __restrict__ sq,
        const int*   __restrict__ tgt,
        float* __restrict__ d_ap,
        float* __restrict__ d_an) {
    if (blockIdx.x < blockIdx.y) return;   // symmetric: skip lower triangle

    const int lane = threadIdx.x;
    const int sub  = lane & 15;
    const int half = lane >> 4;
    const int r0 = blockIdx.y * 64;
    const int c0 = blockIdx.x * 64;

    v8f acc[4][4];
#pragma unroll
    for (int i = 0; i < 4; ++i)
#pragma unroll
        for (int j = 0; j < 4; ++j)
            acc[i][j] = (v8f){};

    const float* Ar[4];
    const float* Br[4];
#pragma unroll
    for (int i = 0; i < 4; ++i)
        Ar[i] = X + (size_t)(r0 + 16 * i + sub) * NDIM + 2 * half;
#pragma unroll
    for (int j = 0; j < 4; ++j)
        Br[j] = X + (size_t)(c0 + 16 * j + sub) * NDIM + 2 * half;

    for (int k = 0; k < NDIM; k += 4) {
        v2f a[4], b[4];
#pragma unroll
        for (int i = 0; i < 4; ++i) a[i] = *(const v2f*)(Ar[i] + k);
#pragma unroll
        for (int j = 0; j < 4; ++j) b[j] = *(const v2f*)(Br[j] + k);
#pragma unroll
        for (int i = 0; i < 4; ++i)
#pragma unroll
            for (int j = 0; j < 4; ++j)
                acc[i][j] = __builtin_amdgcn_wmma_f32_16x16x4_f32(
                    false, a[i], false, b[j], (short)0, acc[i][j], false, false);
    }

    // ---------------- Epilogue: dist + masked hard-mining, both directions.
    const int colbase = c0 + sub;
    float sqc[4];
    int   tc[4];
    float apc[4], anc[4];
#pragma unroll
    for (int j = 0; j < 4; ++j) {
        const int gc = colbase + 16 * j;
        sqc[j] = sq[gc];
        tc[j]  = tgt[gc];
        apc[j] = 0.0f;
        anc[j] = __int_as_float(POS_INF_BITS);
    }

#pragma unroll
    for (int i = 0; i < 4; ++i) {
#pragma unroll
        for (int v = 0; v < 8; ++v) {
            const int gr  = r0 + 16 * i + 8 * half + v;
            const float sqr = sq[gr];
            const int   tr  = tgt[gr];
            float ap = 0.0f;
            float an = __int_as_float(POS_INF_BITS);
#pragma unroll
            for (int j = 0; j < 4; ++j) {
                const float g  = acc[i][j][v];
                const float d2 = sqr + sqc[j] - 2.0f * g;
                const float dist = sqrtf(fmaxf(d2, 1e-12f));
                const bool pos = (tc[j] == tr);
                // row-direction (row gr over these 64 columns)
                ap = pos ? fmaxf(ap, dist) : ap;
                an = pos ? an : fminf(an, dist);
                // column-direction (column gc over these 64 rows)
                apc[j] = pos ? fmaxf(apc[j], dist) : apc[j];
                anc[j] = pos ? anc[j] : fminf(anc[j], dist);
            }
            // reduce row result across the 16 lanes of this half
#pragma unroll
            for (int m = 1; m < 16; m <<= 1) {
                ap = fmaxf(ap, __shfl_xor(ap, m, 32));
                an = fminf(an, __shfl_xor(an, m, 32));
            }
            if (sub == 0) {
                // all values >= 0: int ordering == float ordering
                atomicMax((int*)&d_ap[gr], __float_as_int(ap));
                atomicMin((int*)&d_an[gr], __float_as_int(an));
            }
        }
    }

    // column-direction: combine the two halves (other half holds rows 8..15
    // of each 16-row strip for the same column), then one atomic per column.
#pragma unroll
    for (int j = 0; j < 4; ++j) {
        float apj = fmaxf(apc[j], __shfl_xor(apc[j], 16, 32));
        float anj = fminf(anc[j], __shfl_xor(anc[j], 16, 32));
        if (half == 0) {
            const int gc = colbase + 16 * j;
            atomicMax((int*)&d_ap[gc], __float_as_int(apj));
            atomicMin((int*)&d_an[gc], __float_as_int(anj));
        }
    }
}

// ---------------------------------------------------------------------------
// Kernel 3: final loss / precision reduction.
// ---------------------------------------------------------------------------
__global__ void finalize_kernel(const float* __restrict__ d_ap,
                                const float* __restrict__ d_an,
                                float* __restrict__ out) {
    __shared__ float sl[256];
    __shared__ float sp[256];
    float L = 0.0f, P = 0.0f;
    for (int i = threadIdx.x; i < NROW; i += 256) {
        const float ap = d_ap[i];
        const float an = d_an[i];
        L += fmaxf(ap - an + MARGIN, 0.0f);
        P += (an > ap) ? 1.0f : 0.0f;
    }
    sl[threadIdx.x] = L;
    sp[threadIdx.x] = P;
    __syncthreads();
    for (int off = 128; off > 0; off >>= 1) {
        if (threadIdx.x < off) {
            sl[threadIdx.x] += sl[threadIdx.x + off];
            sp[threadIdx.x] += sp[threadIdx.x + off];
        }
        __syncthreads();
    }
    if (threadIdx.x == 0) {
        out[0] = sl[0] * (1.0f / NROW);
        out[1] = sp[0] * (1.0f / NROW);
    }
}

// ---------------------------------------------------------------------------
extern "C" void kernel_launch(void* const* d_in, const int* in_sizes, int n_in,
                              void* d_out, int out_size, void* d_ws, size_t ws_size,
                              hipStream_t stream) {
    (void)in_sizes; (void)n_in; (void)out_size; (void)ws_size;
    const float* X   = (const float*)d_in[0];
    const int*   tgt = (const int*)d_in[1];
    float* out = (float*)d_out;

    float* sq   = (float*)d_ws;          // 4096 floats
    float* d_ap = sq + NROW;             // 4096 floats
    float* d_an = d_ap + NROW;           // 4096 floats

    prep_kernel<<<NROW, 256, 0, stream>>>(X, sq, d_ap, d_an);

    dim3 grid(NROW / 64, NROW / 64);     // 64x64 tile per wave, upper triangle live
    gram_mine_kernel<<<grid, 32, 0, stream>>>(X, sq, tgt, d_ap, d_an);

    finalize_kernel<<<1, 256, 0, stream>>>(d_ap, d_an, out);
}